// HyperButterflyLayer_54202487275782
// MI455X (gfx1250) — compile-verified
//
#include <hip/hip_runtime.h>

#define DIM     1024
#define PAIRS   (DIM / 2)     // 512 adjacent pairs per row
#define THREADS 256           // 8 wave32s, 4 floats (2 pairs) per thread
#define NWAVES  (THREADS / 32)

typedef float        f32x4 __attribute__((ext_vector_type(4)));
typedef float        f32x2 __attribute__((ext_vector_type(2)));
typedef unsigned int u32x4 __attribute__((ext_vector_type(4)));
typedef int          i32x4 __attribute__((ext_vector_type(4)));
typedef int          i32x8 __attribute__((ext_vector_type(8)));

// ---------------------------------------------------------------------------
// Fold the 4092 butterfly params into one complex multiplier per pair.
// Level l applies z' = (a - i b) z with coeff index j = i >> (l-1); the
// product over l=1..9 is a single complex number (C - iS) per pair i.
// ---------------------------------------------------------------------------
__global__ void hb_precompute_W(const float* __restrict__ params,
                                f32x2* __restrict__ W) {
  int i = blockIdx.x * blockDim.x + threadIdx.x;
  if (i >= PAIRS) return;
  float C = 1.0f, S = 0.0f;
  int off = 2 * DIM;                       // level 0 consumes params but is a no-op
#pragma unroll
  for (int l = 1; l < 10; ++l) {
    int nb = DIM >> l;
    int j  = i >> (l - 1);
    float a = params[off + 2 * j];
    float b = params[off + 2 * j + 1];
    float Cn = C * a - S * b;              // (C - iS)*(a - ib)
    S = S * a + C * b;
    C = Cn;
    off += 2 * nb;
  }
  f32x2 w; w.x = C; w.y = S;
  W[i] = w;
}

// ---------------------------------------------------------------------------
// Main streaming kernel: one block per row.
//   u   = atanh(r)/r * x           (log map at origin, r = ||x||)
//   v   = W (*) u                  (pairwise complex multiply)
//   out = tanh(||v||)/||v|| * v    (exp map at origin, ||v|| clamped 1e-8)
// Row is staged into LDS by the Tensor Data Mover: each wave DMAs its 512B
// slice (1-D tile, 128 x f32) and waits on TENSORcnt.
// ---------------------------------------------------------------------------
__global__ __launch_bounds__(THREADS)
void hb_main(const float* __restrict__ x,
             const f32x4* __restrict__ W,   // (C0,S0,C1,S1) per thread
             float* __restrict__ out) {
  __shared__ float lds_row[DIM];
  __shared__ float red[2][NWAVES];

  const int tid  = threadIdx.x;
  const int lane = tid & 31;
  const int wave = __builtin_amdgcn_readfirstlane(tid >> 5);
  const float* xrow = x + (size_t)blockIdx.x * DIM;

  // ---- TDM stage: per-wave 128-element slice -> LDS ----
  {
    unsigned long long ga = (unsigned long long)(xrow + wave * 128);
    unsigned int       lh = (unsigned int)(size_t)(void*)lds_row + (unsigned)wave * 512u;

    const unsigned int td0 = 128, tile0 = 128, td1 = 1, st0 = 128, st1 = 128;

    u32x4 g0;
    g0.x = 1u;                                               // count=1, user mode
    g0.y = lh;                                               // lds_addr (bytes)
    g0.z = (unsigned int)(ga & 0xffffffffull);               // global_addr[31:0]
    g0.w = (unsigned int)((ga >> 32) & 0x1ffffffull)         // global_addr[56:32]
         | (2u << 30);                                       // type = 2 ("image")

    i32x8 g1;
    g1[0] = (int)(2u << 16);                                 // data_size = 4B
    g1[1] = (int)((td0 & 0xffffu) << 16);                    // tensor_dim0 lo16 @bit48
    g1[2] = (int)((td0 >> 16) | ((td1 & 0xffffu) << 16));    // dim0 hi | dim1 lo
    g1[3] = (int)((td1 >> 16) | (tile0 << 16));              // dim1 hi | tile_dim0
    g1[4] = (int)1;                                          // tile_dim1=1, tile_dim2=0
    g1[5] = (int)st0;                                        // tensor_dim0_stride lo32
    g1[6] = (int)((st1 & 0xffffu) << 16);                    // stride0 hi16 | stride1 lo16
    g1[7] = (int)(st1 >> 16);                                // stride1 hi32
    i32x4 g2 = {0, 0, 0, 0};
    i32x4 g3 = {0, 0, 0, 0};
#if defined(__clang_major__) && (__clang_major__ >= 23)
    i32x8 g4 = {0, 0, 0, 0, 0, 0, 0, 0};
    __builtin_amdgcn_tensor_load_to_lds(g0, g1, g2, g3, g4, 0);
#else
    __builtin_amdgcn_tensor_load_to_lds(g0, g1, g2, g3, 0);
#endif
  }
  __builtin_amdgcn_s_wait_tensorcnt(0);
  __syncthreads();

  // ---- per-thread: 2 pairs, complex rotate, partial norms ----
  f32x4 xv = *reinterpret_cast<const f32x4*>(lds_row + 4 * tid); // ds_load_b128
  f32x4 wv = W[tid];

  f32x4 v;
  v.x =  wv.x * xv.x + wv.y * xv.y;
  v.y = -wv.y * xv.x + wv.x * xv.y;
  v.z =  wv.z * xv.z + wv.w * xv.w;
  v.w = -wv.w * xv.z + wv.z * xv.w;

  float sx = xv.x * xv.x + xv.y * xv.y + xv.z * xv.z + xv.w * xv.w;
  float sv = v.x * v.x + v.y * v.y + v.z * v.z + v.w * v.w;

  // ---- wave32 butterfly reduction, then cross-wave via LDS ----
#pragma unroll
  for (int m = 16; m > 0; m >>= 1) {
    sx += __shfl_xor(sx, m, 32);
    sv += __shfl_xor(sv, m, 32);
  }
  if (lane == 0) { red[0][wave] = sx; red[1][wave] = sv; }
  __syncthreads();
  sx = 0.0f; sv = 0.0f;
#pragma unroll
  for (int w = 0; w < NWAVES; ++w) { sx += red[0][w]; sv += red[1][w]; }

  // ---- radial scales (log map then exp map at origin) ----
  float r     = sqrtf(sx);
  float s_in  = atanhf(r) / r;            // matches reference 0/0 behavior
  float vn    = s_in * sqrtf(sv);         // ||v|| since v scales uniformly
  vn          = fmaxf(vn, 1e-8f);
  float scale = s_in * tanhf(vn) / vn;

  f32x4 o;
  o.x = scale * v.x; o.y = scale * v.y; o.z = scale * v.z; o.w = scale * v.w;
  __builtin_nontemporal_store(
      o, reinterpret_cast<f32x4*>(out + (size_t)blockIdx.x * DIM + 4 * tid));
}

// ---------------------------------------------------------------------------
extern "C" void kernel_launch(void* const* d_in, const int* in_sizes, int n_in,
                              void* d_out, int out_size, void* d_ws, size_t ws_size,
                              hipStream_t stream) {
  const float* x      = (const float*)d_in[0];
  const float* params = (const float*)d_in[1];
  float*       out    = (float*)d_out;
  f32x2*       W      = (f32x2*)d_ws;            // 512 * 8B = 4 KiB scratch

  const int rows = in_sizes[0] / DIM;            // 32768

  hb_precompute_W<<<(PAIRS + 255) / 256, 256, 0, stream>>>(params, W);
  hb_main<<<rows, THREADS, 0, stream>>>(x, (const f32x4*)d_ws, out);
}